// ChebNet_6665789243575
// MI455X (gfx1250) — compile-verified
//
#include <hip/hip_runtime.h>
#include <hip/hip_bf16.h>

typedef float v2f __attribute__((ext_vector_type(2)));
typedef float v8f __attribute__((ext_vector_type(8)));

#define BN_EPS 1e-5f

// ---------------- graph preprocessing ----------------

__global__ void k_degree(const int* __restrict__ row, const float* __restrict__ ea,
                         float* __restrict__ deg, int E) {
    int e = blockIdx.x * blockDim.x + threadIdx.x;
    if (e < E) atomicAdd(&deg[row[e]], ea[e]);
}

__global__ void k_dinv(float* __restrict__ deg, int N) {
    int i = blockIdx.x * blockDim.x + threadIdx.x;
    if (i < N) {
        float d = deg[i];
        deg[i] = (d > 0.f) ? rsqrtf(d) : 0.f;
    }
}

__global__ void k_norm(const int* __restrict__ row, const int* __restrict__ col,
                       const float* __restrict__ ea, const float* __restrict__ dinv,
                       float* __restrict__ nrm, int E) {
    int e = blockIdx.x * blockDim.x + threadIdx.x;
    if (e < E) nrm[e] = -ea[e] * dinv[row[e]] * dinv[col[e]];
}

// ---------------- weight repack into WMMA-native B layout ----------------
// Wp[s][kblk][half][n][j] = W[s][kblk*4 + half*2 + j][n], zero-padded to
// Kpad (mult of 4) x Npad (mult of 16). Lane (n, kb=half*2) then reads its
// (b.x, b.y) pair as one aligned 8-byte load.

__global__ void k_pack(const float* __restrict__ W, float* __restrict__ Wp,
                       int Fin, int Fout, int Kpad, int Npad) {
    int t = blockIdx.x * blockDim.x + threadIdx.x;
    int per = Kpad * Npad;                 // floats per s
    if (t >= 2 * per) return;
    int s = t / per, r = t % per;
    int j = r & 1;
    int n = (r >> 1) % Npad;
    int hh = (r >> 1) / Npad;              // = kblk*2 + half
    int k = (hh >> 1) * 4 + (hh & 1) * 2 + j;
    Wp[t] = (k < Fin && n < Fout) ? W[(long)s * Fin * Fout + (long)k * Fout + n] : 0.f;
}

// ---------------- SpMM scatter: tx1[row] += norm * h[col] ----------------
// consecutive threads cover consecutive 4-feature chunks of one edge -> coalesced

__global__ void k_spmm(const int* __restrict__ row, const int* __restrict__ col,
                       const float* __restrict__ nrm, const float* __restrict__ h,
                       float* __restrict__ tx1, int E, int F, int nCh) {
    long tid = (long)blockIdx.x * blockDim.x + threadIdx.x;
    long total = (long)E * nCh;
    if (tid >= total) return;
    int e  = (int)(tid / nCh);
    int c  = (int)(tid % nCh);
    int f0 = c * 4;
    float w = nrm[e];
    if (w == 0.f) return;
    const float* src = h   + (long)col[e] * F + f0;
    float*       dst = tx1 + (long)row[e] * F + f0;
#pragma unroll
    for (int i = 0; i < 4; ++i) {
        int f = f0 + i;
        if (f < F) atomicAdd(dst + i, w * src[i]);
    }
}

// ---------------- fused Cheb GEMM: out = BN(relu(h@W0 + tx1@W1 + b)) + alpha*res ----
// one wave per 16x16 output tile; K-steps of 4 via V_WMMA_F32_16X16X4_F32.
// B comes pre-packed (zero-padded), so the main loop is 2 aligned b64 loads
// per WMMA with no guards; only the TAIL instantiation has one masked A step.

template <bool TAIL>
__global__ __launch_bounds__(128)
void k_cheb(const float* __restrict__ h, const float* __restrict__ tx1,
            const float* __restrict__ Wp, const float* __restrict__ bias,
            const float* __restrict__ bng, const float* __restrict__ bnb,
            const float* __restrict__ bnm, const float* __restrict__ bnv,
            const float* __restrict__ res, float alpha, float* __restrict__ out,
            int N, int Fin, int Fout, int Kpad, int Npad) {
    int lane = threadIdx.x & 31;
    int wid  = (blockIdx.x * blockDim.x + threadIdx.x) >> 5;
    int colTiles = Npad >> 4;
    int rowTiles = N >> 4;                       // N is a multiple of 16
    if (wid >= rowTiles * colTiles) return;
    // waves of a block share the column tile (B reuse), adjacent row tiles
    int tn = wid / rowTiles, tm = wid % rowTiles;

    int mrow = tm * 16 + (lane & 15);            // A row held by this lane
    int ncol = tn * 16 + (lane & 15);            // B/D column held by this lane
    int kb   = (lane >> 4) * 2;                  // k sub-offset within 4-step

    const int  Kfull   = Fin >> 2;               // unguarded 4-k blocks
    const long sStride = (long)Kpad * Npad;
    const long bStep   = 4 * (long)Npad;

    v8f acc = {};
    const float* Src[2] = { h, tx1 };
    const float* bp0 = Wp + (((long)(lane >> 4)) * Npad + ncol) * 2;

#pragma unroll
    for (int s = 0; s < 2; ++s) {
        const float* arow = Src[s] + (long)mrow * Fin;
        const float* bp   = bp0 + s * sStride;
#pragma unroll 4
        for (int kblk = 0; kblk < Kfull; ++kblk) {
            v2f a, b;
            if (TAIL) {                           // rows only 4B-aligned
                a.x = arow[kblk * 4 + kb];
                a.y = arow[kblk * 4 + kb + 1];
            } else {                              // Fin % 4 == 0 -> 8B-aligned
                a = *(const v2f*)(arow + kblk * 4 + kb);
            }
            b = *(const v2f*)bp;
            bp += bStep;
            acc = __builtin_amdgcn_wmma_f32_16x16x4_f32(false, a, false, b,
                                                        (short)0, acc, false, false);
        }
        if (TAIL && (Fin & 3)) {                  // one branchless masked A step
            int k0 = Kfull * 4 + kb, k1 = k0 + 1;
            float m0 = (k0 < Fin) ? 1.f : 0.f;
            float m1 = (k1 < Fin) ? 1.f : 0.f;
            int  i0 = (k0 < Fin) ? k0 : (Fin - 1);
            int  i1 = (k1 < Fin) ? k1 : (Fin - 1);
            v2f a, b;
            a.x = arow[i0] * m0;
            a.y = arow[i1] * m1;
            b = *(const v2f*)bp;                  // padded region is zeros
            acc = __builtin_amdgcn_wmma_f32_16x16x4_f32(false, a, false, b,
                                                        (short)0, acc, false, false);
        }
    }

    if (ncol < Fout) {
        float bi = bias[ncol];
        float sc = bng[ncol] * rsqrtf(bnv[ncol] + BN_EPS);
        float sh = bnb[ncol] - bnm[ncol] * sc;
        int r0 = tm * 16 + ((lane >> 4) * 8);
#pragma unroll
        for (int i = 0; i < 8; ++i) {
            int r = r0 + i;
            float y = acc[i] + bi;
            y = fmaxf(y, 0.f);
            y = y * sc + sh;
            if (res) y += alpha * res[(long)r * Fout + ncol];
            out[(long)r * Fout + ncol] = y;
        }
    }
}

// ---------------- FC readout: out[g][j] = sum_k A[g][k]*fcW[j][k] + fcb[j] --------

__global__ __launch_bounds__(128)
void k_fc(const float* __restrict__ A, const float* __restrict__ fcW,
          const float* __restrict__ fcb, float* __restrict__ out,
          int G, int K, int J) {
    int lane = threadIdx.x & 31;
    int wid  = (blockIdx.x * blockDim.x + threadIdx.x) >> 5;
    int colTiles = J >> 4, rowTiles = G >> 4;
    if (wid >= rowTiles * colTiles) return;
    int tn = wid / rowTiles, tm = wid % rowTiles;

    int mrow = tm * 16 + (lane & 15);
    int ncol = tn * 16 + (lane & 15);
    int kb   = (lane >> 4) * 2;

    const float* arow = A + (long)mrow * K;
    const float* brow = fcW + (long)ncol * K;    // B[k][n] = fcW[n*K + k]

    v8f acc = {};
#pragma unroll 4
    for (int k = 0; k < K; k += 4) {             // K = 1656 is a multiple of 4
        v2f a, b;
        a.x = arow[k + kb];  a.y = arow[k + kb + 1];
        b.x = brow[k + kb];  b.y = brow[k + kb + 1];
        acc = __builtin_amdgcn_wmma_f32_16x16x4_f32(false, a, false, b,
                                                    (short)0, acc, false, false);
    }

    float bi = fcb[ncol];
    int r0 = tm * 16 + ((lane >> 4) * 8);
#pragma unroll
    for (int i = 0; i < 8; ++i)
        out[(long)(r0 + i) * J + ncol] = acc[i] + bi;
}

// ---------------- host orchestration ----------------

static inline long cdiv(long a, long b) { return (a + b - 1) / b; }

extern "C" void kernel_launch(void* const* d_in, const int* in_sizes, int n_in,
                              void* d_out, int out_size, void* d_ws, size_t ws_size,
                              hipStream_t stream) {
    const long N = in_sizes[0] / 207;            // 26496
    const long E = in_sizes[2];                  // 847872

    const float* x    = (const float*)d_in[0];
    const int*   ei   = (const int*)d_in[1];
    const int*   row  = ei;
    const int*   colI = ei + E;
    const float* ea   = (const float*)d_in[2];
    // params in setup_inputs() dict insertion order
    const float* W1   = (const float*)d_in[4];
    const float* b1   = (const float*)d_in[5];
    const float* W2_0 = (const float*)d_in[6];
    const float* W2_1 = (const float*)d_in[7];
    const float* W2_2 = (const float*)d_in[8];
    const float* b2_0 = (const float*)d_in[9];
    const float* b2_1 = (const float*)d_in[10];
    const float* b2_2 = (const float*)d_in[11];
    const float* W3   = (const float*)d_in[12];
    const float* b3   = (const float*)d_in[13];
    const float* W4   = (const float*)d_in[14];
    const float* b4   = (const float*)d_in[15];
    const float* bn1g = (const float*)d_in[16], *bn1b = (const float*)d_in[17],
               * bn1m = (const float*)d_in[18], *bn1v = (const float*)d_in[19];
    const float* bn2g = (const float*)d_in[20], *bn2b = (const float*)d_in[21],
               * bn2m = (const float*)d_in[22], *bn2v = (const float*)d_in[23];
    const float* bn3g = (const float*)d_in[24], *bn3b = (const float*)d_in[25],
               * bn3m = (const float*)d_in[26], *bn3v = (const float*)d_in[27];
    const float* fcW  = (const float*)d_in[28];
    const float* fcb  = (const float*)d_in[29];

    // workspace layout (all offsets even -> 8B alignment preserved)
    float* ws  = (float*)d_ws;
    float* deg = ws;              long off = N;
    float* nrm = ws + off;        off += E;
    float* tx1 = ws + off;        off += N * 207;
    float* hA  = ws + off;        off += N * 128;
    float* hB  = ws + off;        off += N * 128;
    float* h8  = ws + off;        off += N * 8;
    float* Wp1  = ws + off;       off += 2 * 208 * 128;
    float* Wp20 = ws + off;       off += 2 * 128 * 128;
    float* Wp21 = ws + off;       off += 2 * 128 * 128;
    float* Wp22 = ws + off;       off += 2 * 128 * 128;
    float* Wp3  = ws + off;       off += 2 * 128 * 64;
    float* Wp4  = ws + off;       off += 2 * 64 * 16;

    // 0) repack weights into WMMA-native padded B layout
    auto pack = [&](const float* W, float* Wp, int Fin, int Fout, int Kpad, int Npad) {
        long tot = 2L * Kpad * Npad;
        k_pack<<<cdiv(tot, 256), 256, 0, stream>>>(W, Wp, Fin, Fout, Kpad, Npad);
    };
    pack(W1,   Wp1,  207, 128, 208, 128);
    pack(W2_0, Wp20, 128, 128, 128, 128);
    pack(W2_1, Wp21, 128, 128, 128, 128);
    pack(W2_2, Wp22, 128, 128, 128, 128);
    pack(W3,   Wp3,  128,  64, 128,  64);
    pack(W4,   Wp4,   64,   8,  64,  16);

    // 1) normalization coefficients
    hipMemsetAsync(deg, 0, N * sizeof(float), stream);
    k_degree<<<cdiv(E, 256), 256, 0, stream>>>(row, ea, deg, (int)E);
    k_dinv  <<<cdiv(N, 256), 256, 0, stream>>>(deg, (int)N);
    k_norm  <<<cdiv(E, 256), 256, 0, stream>>>(row, colI, ea, deg, nrm, (int)E);

    auto spmm = [&](const float* h, int F) {
        hipMemsetAsync(tx1, 0, (size_t)N * F * sizeof(float), stream);
        int nCh = (F + 3) / 4;
        long tot = E * (long)nCh;
        k_spmm<<<cdiv(tot, 256), 256, 0, stream>>>(row, colI, nrm, h, tx1, (int)E, F, nCh);
    };
    auto cheb = [&](const float* h, const float* Wp, const float* bias,
                    const float* g, const float* bb, const float* m, const float* v,
                    const float* res, float alpha, float* out,
                    int Fin, int Fout, int Kpad, int Npad, bool tail) {
        long waves = (N / 16) * (long)(Npad / 16);
        long blocks = cdiv(waves, 4);
        if (tail)
            k_cheb<true><<<blocks, 128, 0, stream>>>(h, tx1, Wp, bias, g, bb, m, v,
                                                     res, alpha, out, (int)N, Fin, Fout, Kpad, Npad);
        else
            k_cheb<false><<<blocks, 128, 0, stream>>>(h, tx1, Wp, bias, g, bb, m, v,
                                                      res, alpha, out, (int)N, Fin, Fout, Kpad, Npad);
    };

    // 2) layer 1: x(207) -> hA(128)
    spmm(x, 207);
    cheb(x, Wp1, b1, bn1g, bn1b, bn1m, bn1v, nullptr, 0.f, hA, 207, 128, 208, 128, true);
    // 3) residual blocks (128 -> 128), res = layer input, alpha = 0.7
    spmm(hA, 128);
    cheb(hA, Wp20, b2_0, bn1g, bn1b, bn1m, bn1v, hA, 0.7f, hB, 128, 128, 128, 128, false);
    spmm(hB, 128);
    cheb(hB, Wp21, b2_1, bn1g, bn1b, bn1m, bn1v, hB, 0.7f, hA, 128, 128, 128, 128, false);
    spmm(hA, 128);
    cheb(hA, Wp22, b2_2, bn1g, bn1b, bn1m, bn1v, hA, 0.7f, hB, 128, 128, 128, 128, false);
    // 4) layer 3: 128 -> 64 (into hA)
    spmm(hB, 128);
    cheb(hB, Wp3, b3, bn2g, bn2b, bn2m, bn2v, nullptr, 0.f, hA, 128, 64, 128, 64, false);
    // 5) layer 4: 64 -> 8 (into h8)
    spmm(hA, 64);
    cheb(hA, Wp4, b4, bn3g, bn3b, bn3m, bn3v, nullptr, 0.f, h8, 64, 8, 64, 16, false);

    // 6) FC readout: [G=128 x 1656] @ fcW^T + fcb -> d_out [128 x 128]
    const int G = (int)(N / 207), K = 207 * 8, J = 128;
    long fcWaves = (G / 16) * (long)(J / 16);
    k_fc<<<cdiv(fcWaves, 4), 128, 0, stream>>>(h8, fcW, fcb, (float*)d_out, G, K, J);
}